// MHAHead_68994354642948
// MI455X (gfx1250) — compile-verified
//
#include <hip/hip_runtime.h>
#include <math.h>

typedef __attribute__((ext_vector_type(2))) float v2f;
typedef __attribute__((ext_vector_type(4))) float v4f;
typedef __attribute__((ext_vector_type(8))) float v8f;

#define EMB 512
#define RNK 64
#define BB  8
#define SS  2048
#define NROWS (BB*SS)          // 16384

// ---------------------------------------------------------------------------
// Generic fp32 WMMA GEMM: C = op(A) @ B (+bias), optional per-row int mask:
//   C = (acc + bias) * rowscale * (rowmask ? mask[row] : 1)
// op(A)=A (TA=false, row-major MxK, lda) or A^T (TA=true, A row-major KxM).
// B row-major KxN (ldb), C row-major MxN (ldc).
// Each wave computes a (16*MT) x (16*NT) block of C with
// V_WMMA_F32_16X16X4_F32; B fragments are reused across MT M-tiles and
// A fragments across NT N-tiles (10 VMEM / 8 WMMA at MT=2,NT=4).
// blockIdx.y = batch; sA/sB/sC are per-batch element strides.
// ---------------------------------------------------------------------------
template <int MT, int NT, bool TA>
__global__ __launch_bounds__(256)
void gemm_wmma_f32(const float* __restrict__ A, const float* __restrict__ B,
                   float* __restrict__ C, const float* __restrict__ bias,
                   const int* __restrict__ rowmask, float rowscale,
                   int M, int N, int K, int lda, int ldb, int ldc,
                   long sA, long sB, long sC, int maskStride)
{
    const int batch = blockIdx.y;
    A += (long)batch * sA;
    B += (long)batch * sB;
    C += (long)batch * sC;
    const int* rmask = rowmask ? (rowmask + (long)batch * maskStride) : nullptr;

    const int lane = threadIdx.x & 31;
    const int wave = threadIdx.x >> 5;
    const int l    = lane & 15;     // M index (A), N index (B/C)
    const int half = lane >> 4;     // K sub-block select (K = 2*half + vgpr)

    const int mstrips = M / (16 * MT);
    const int nstrips = N / (16 * NT);
    const int nTotal  = mstrips * nstrips;

    const int strip = blockIdx.x * (blockDim.x >> 5) + wave;
    if (strip >= nTotal) return;

    const int mr  = (strip / nstrips) * (16 * MT);
    const int nc0 = (strip % nstrips) * (16 * NT);

    v8f acc[MT][NT] = {};

#pragma unroll 2
    for (int kb = 0; kb < K; kb += 4) {
        // A fragments: 16x4 tiles, lanes = M, K = 2*half + {vgpr0, vgpr1}
        v2f a[MT];
#pragma unroll
        for (int mt = 0; mt < MT; ++mt) {
            const int mrow = mr + 16 * mt + l;
            if (TA) {
                a[mt][0] = A[(long)(kb + 2 * half)     * lda + mrow];
                a[mt][1] = A[(long)(kb + 2 * half + 1) * lda + mrow];
            } else {
                a[mt] = *(const v2f*)&A[(long)mrow * lda + kb + 2 * half];
            }
        }
#pragma unroll
        for (int t = 0; t < NT; ++t) {
            const int nc = nc0 + 16 * t;
            // B fragment: 4x16, lanes = N, K = 2*half + {vgpr0, vgpr1}
            v2f b;
            b[0] = B[(long)(kb + 2 * half)     * ldb + nc + l];
            b[1] = B[(long)(kb + 2 * half + 1) * ldb + nc + l];
#pragma unroll
            for (int mt = 0; mt < MT; ++mt) {
                acc[mt][t] = __builtin_amdgcn_wmma_f32_16x16x4_f32(
                    false, a[mt], false, b, (short)0, acc[mt][t], false, false);
            }
        }
    }

    // Epilogue. D layout: VGPR i holds row M = i + 8*half, lane = N.
#pragma unroll
    for (int mt = 0; mt < MT; ++mt) {
        float f[8];
#pragma unroll
        for (int i = 0; i < 8; ++i) {
            const int row = mr + 16 * mt + i + 8 * half;
            f[i] = rmask ? rowscale * (float)rmask[row] : rowscale;
        }
#pragma unroll
        for (int t = 0; t < NT; ++t) {
            const int nc = nc0 + 16 * t;
            const float bv = bias ? bias[nc + l] : 0.0f;
#pragma unroll
            for (int i = 0; i < 8; ++i) {
                const int row = mr + 16 * mt + i + 8 * half;
                C[(long)row * ldc + nc + l] = (acc[mt][t][i] + bv) * f[i];
            }
        }
    }
}

// ---------------------------------------------------------------------------
// Row softmax over 512 columns, in place. One wave per row; 4 x b128 per lane.
// ---------------------------------------------------------------------------
__global__ __launch_bounds__(256)
void softmax512_rows(float* __restrict__ data, int nrows)
{
    const int lane = threadIdx.x & 31;
    const int wave = threadIdx.x >> 5;
    const int row  = blockIdx.x * 8 + wave;
    if (row >= nrows) return;

    v4f* p = (v4f*)(data + (long)row * EMB);   // 128 v4f per row
    v4f v[4];
    float mx = -INFINITY;
#pragma unroll
    for (int i = 0; i < 4; ++i) {
        v[i] = p[lane + 32 * i];
#pragma unroll
        for (int j = 0; j < 4; ++j) mx = fmaxf(mx, v[i][j]);
    }
#pragma unroll
    for (int off = 16; off > 0; off >>= 1)
        mx = fmaxf(mx, __shfl_xor(mx, off, 32));

    float sum = 0.0f;
#pragma unroll
    for (int i = 0; i < 4; ++i)
#pragma unroll
        for (int j = 0; j < 4; ++j) {
            v[i][j] = __expf(v[i][j] - mx);
            sum += v[i][j];
        }
#pragma unroll
    for (int off = 16; off > 0; off >>= 1)
        sum += __shfl_xor(sum, off, 32);

    const float inv = 1.0f / sum;
#pragma unroll
    for (int i = 0; i < 4; ++i) {
#pragma unroll
        for (int j = 0; j < 4; ++j) v[i][j] *= inv;
        p[lane + 32 * i] = v[i];
    }
}

// ---------------------------------------------------------------------------
extern "C" void kernel_launch(void* const* d_in, const int* in_sizes, int n_in,
                              void* d_out, int out_size, void* d_ws, size_t ws_size,
                              hipStream_t stream)
{
    (void)in_sizes; (void)n_in; (void)out_size; (void)ws_size;

    const float* x    = (const float*)d_in[0];
    const int*   mask = (const int*)  d_in[1];
    const float* q_w1 = (const float*)d_in[2];
    const float* q_b1 = (const float*)d_in[3];
    const float* q_w2 = (const float*)d_in[4];
    const float* q_b2 = (const float*)d_in[5];
    const float* k_w1 = (const float*)d_in[6];
    const float* k_b1 = (const float*)d_in[7];
    const float* k_w2 = (const float*)d_in[8];
    const float* k_b2 = (const float*)d_in[9];
    const float* v_w1 = (const float*)d_in[10];
    const float* v_b1 = (const float*)d_in[11];
    const float* v_w2 = (const float*)d_in[12];
    const float* v_b2 = (const float*)d_in[13];

    float* ws = (float*)d_ws;
    float* Hq = ws;                         // [16384 x 64]
    float* Hk = Hq + (long)NROWS * RNK;
    float* Hv = Hk + (long)NROWS * RNK;
    float* Km = Hv + (long)NROWS * RNK;     // masked K  [16384 x 512]
    float* Vp = Km + (long)NROWS * EMB;     // V         [16384 x 512]
    float* Mm = Vp + (long)NROWS * EMB;     // per-batch K_m^T V [8 x 512 x 512]

    float* out = (float*)d_out;             // softmax(temp)  [16384 x 512]
    float* Qp  = out + (long)NROWS * EMB;   // returned q     [16384 x 512]

    const dim3 blk(256);
    const float scale = 1.0f / sqrtf((float)EMB);

    // ---- Stage 1: H_h = X @ W1_h + b1_h   (M=16384, N=64, K=512) ----------
    {
        dim3 grd((NROWS / 32) / 8, 1);      // 512 strips (nstrips = 1)
        gemm_wmma_f32<2, 4, false><<<grd, blk, 0, stream>>>(
            x, q_w1, Hq, q_b1, nullptr, 1.0f,
            NROWS, RNK, EMB, EMB, RNK, RNK, 0, 0, 0, 0);
        gemm_wmma_f32<2, 4, false><<<grd, blk, 0, stream>>>(
            x, k_w1, Hk, k_b1, nullptr, 1.0f,
            NROWS, RNK, EMB, EMB, RNK, RNK, 0, 0, 0, 0);
        gemm_wmma_f32<2, 4, false><<<grd, blk, 0, stream>>>(
            x, v_w1, Hv, v_b1, nullptr, 1.0f,
            NROWS, RNK, EMB, EMB, RNK, RNK, 0, 0, 0, 0);
    }

    // ---- Stage 2: P_h = H_h @ W2_h + b2_h (M=16384, N=512, K=64) ----------
    {
        dim3 grd(((NROWS / 32) * (EMB / 64)) / 8, 1);   // 4096/8 = 512 blocks
        // Q -> straight into d_out's second return slot (it is 'q')
        gemm_wmma_f32<2, 4, false><<<grd, blk, 0, stream>>>(
            Hq, q_w2, Qp, q_b2, nullptr, 1.0f,
            NROWS, EMB, RNK, RNK, EMB, EMB, 0, 0, 0, 0);
        // K_m = diag(mask) * (H_k @ W2 + b2): fold mask at the epilogue
        gemm_wmma_f32<2, 4, false><<<grd, blk, 0, stream>>>(
            Hk, k_w2, Km, k_b2, mask, 1.0f,
            NROWS, EMB, RNK, RNK, EMB, EMB, 0, 0, 0, NROWS /*stride unused, batch=0*/);
        gemm_wmma_f32<2, 4, false><<<grd, blk, 0, stream>>>(
            Hv, v_w2, Vp, v_b2, nullptr, 1.0f,
            NROWS, EMB, RNK, RNK, EMB, EMB, 0, 0, 0, 0);
    }

    // ---- Stage 3: M_b = K_m^T @ V  per batch (M=N=512, K=2048) ------------
    {
        dim3 grd(((EMB / 32) * (EMB / 64)) / 8, BB);    // 128/8 = 16, y = 8
        gemm_wmma_f32<2, 4, true><<<grd, blk, 0, stream>>>(
            Km, Vp, Mm, nullptr, nullptr, 1.0f,
            EMB, EMB, SS, EMB, EMB, EMB,
            (long)SS * EMB, (long)SS * EMB, (long)EMB * EMB, 0);
    }

    // ---- Stage 4: temp = scale * m[s] * (Q_b @ M_b)  (M=2048,N=512,K=512) -
    {
        dim3 grd(((SS / 32) * (EMB / 64)) / 8, BB);     // 512/8 = 64, y = 8
        gemm_wmma_f32<2, 4, false><<<grd, blk, 0, stream>>>(
            Qp, Mm, out, nullptr, mask, scale,
            SS, EMB, EMB, EMB, EMB, EMB,
            (long)SS * EMB, (long)EMB * EMB, (long)SS * EMB, SS);
    }

    // ---- Stage 5: softmax over E (in place on d_out) ----------------------
    softmax512_rows<<<NROWS / 8, blk, 0, stream>>>(out, NROWS);
}